// GNNBackBone_78391743086995
// MI455X (gfx1250) — compile-verified
//
#include <hip/hip_runtime.h>

typedef float v2f __attribute__((ext_vector_type(2)));
typedef float v8f __attribute__((ext_vector_type(8)));

#define DIM 32

// ---------------- utility / sparse kernels ----------------

__global__ void zero_f32_kernel(float* __restrict__ p, int n) {
  int i = blockIdx.x * blockDim.x + threadIdx.x;
  if (i < n) p[i] = 0.0f;
}

__global__ void deg_kernel(const int* __restrict__ col, const float* __restrict__ w,
                           float* __restrict__ deg, int E) {
  int e = blockIdx.x * blockDim.x + threadIdx.x;
  if (e < E) atomicAdd(&deg[col[e]], w[e]);
}

__global__ void dis_kernel(const float* __restrict__ deg, float* __restrict__ dis, int N) {
  int i = blockIdx.x * blockDim.x + threadIdx.x;
  if (i < N) {
    float d = deg[i];
    dis[i] = (d > 0.0f) ? rsqrtf(d) : 0.0f;
  }
}

__global__ void norm_kernel(const int* __restrict__ row, const int* __restrict__ col,
                            const float* __restrict__ w, const float* __restrict__ dis,
                            float* __restrict__ nrm, int E) {
  int e = blockIdx.x * blockDim.x + threadIdx.x;
  if (e < E) nrm[e] = dis[row[e]] * w[e] * dis[col[e]];
}

// One 32-lane group per edge: lane d handles feature dim d.
// row/col/norm are uniform within the group (scalarized); the h row gather and
// the atomic scatter are each a coalesced 128B burst, L2-resident on MI455X.
__global__ void hop_kernel(const int* __restrict__ row, const int* __restrict__ col,
                           const float* __restrict__ nrm,
                           const float* __restrict__ hsrc, float* __restrict__ hdst, int E) {
  int t = blockIdx.x * blockDim.x + threadIdx.x;
  int e = t >> 5;
  int d = t & 31;
  if (e >= E) return;
  int r = row[e];
  int c = col[e];
  float v = nrm[e] * hsrc[(long)r * DIM + d];
  atomicAdd(&hdst[(long)c * DIM + d], v);
}

// ---------------- fused 4-way GEMM + bias + ReLU (WMMA f32 16x16x4) ----------------
// dst = relu(bias + x0@W[0] + h1@W[1] + h2@W[2] + h3@W[3])
// One wave per 16-row tile; two 16x16 f32 accumulators cover the 32 output cols.

__global__ __launch_bounds__(256)
void tag_gemm_wmma_kernel(const float* __restrict__ s0, const float* __restrict__ s1,
                          const float* __restrict__ s2, const float* __restrict__ s3,
                          const float* __restrict__ Wl,   // [4][32][32] row-major (j,n)
                          const float* __restrict__ bl,   // [32]
                          float* __restrict__ dst, int nTiles) {
  int wave = threadIdx.x >> 5;
  int tile = blockIdx.x * (blockDim.x >> 5) + wave;
  if (tile >= nTiles) return;          // whole-wave uniform exit: EXEC stays all-1s
  int lane = threadIdx.x & 31;
  int mlane = lane & 15;
  bool hi = lane >= 16;
  int khalf = hi ? 2 : 0;
  long rowBase = (long)(tile * 16 + mlane) * DIM;

  v8f c0 = {};  // output cols 0..15
  v8f c1 = {};  // output cols 16..31

  const float* srcs[4] = {s0, s1, s2, s3};
#pragma unroll
  for (int k = 0; k < 4; ++k) {
    const float* S = srcs[k];
    const float* Wk = Wl + k * (DIM * DIM);
#pragma unroll
    for (int kk = 0; kk < DIM; kk += 4) {
      int ka = kk + khalf;
      // A 16x4 fp32 layout: lane<16 -> (K=kk,kk+1), lane>=16 -> (K=kk+2,kk+3); M = lane&15
      v2f a = *reinterpret_cast<const v2f*>(S + rowBase + ka);
      // B 4x16 layout: VGPR0 lanes0-15 K=kk / lanes16-31 K=kk+2; VGPR1 K=kk+1 / kk+3
      v2f b0, b1;
      b0.x = Wk[(ka + 0) * DIM + mlane];
      b0.y = Wk[(ka + 1) * DIM + mlane];
      b1.x = Wk[(ka + 0) * DIM + mlane + 16];
      b1.y = Wk[(ka + 1) * DIM + mlane + 16];
      c0 = __builtin_amdgcn_wmma_f32_16x16x4_f32(false, a, false, b0, (short)0, c0, false, false);
      c1 = __builtin_amdgcn_wmma_f32_16x16x4_f32(false, a, false, b1, (short)0, c1, false, false);
    }
  }

  // C 16x16 f32 layout: VGPR r -> row r (lanes 0-15) / row r+8 (lanes 16-31), N = lane&15
  float bn0 = bl[mlane];
  float bn1 = bl[mlane + 16];
  int mBase = tile * 16 + (hi ? 8 : 0);
#pragma unroll
  for (int r = 0; r < 8; ++r) {
    float v0 = c0[r] + bn0;
    float v1 = c1[r] + bn1;
    v0 = v0 > 0.0f ? v0 : 0.0f;
    v1 = v1 > 0.0f ? v1 : 0.0f;
    dst[(long)(mBase + r) * DIM + mlane]      = v0;
    dst[(long)(mBase + r) * DIM + mlane + 16] = v1;
  }
}

// Scalar tail for N % 16 rows (never launched for N=100000, kept for generality).
__global__ void tag_gemm_tail_kernel(const float* __restrict__ s0, const float* __restrict__ s1,
                                     const float* __restrict__ s2, const float* __restrict__ s3,
                                     const float* __restrict__ Wl, const float* __restrict__ bl,
                                     float* __restrict__ dst, int startRow, int N) {
  int t = blockIdx.x * blockDim.x + threadIdx.x;
  int m = startRow + (t >> 5);
  int n = t & 31;
  if (m >= N) return;
  float acc = bl[n];
  const float* srcs[4] = {s0, s1, s2, s3};
  for (int k = 0; k < 4; ++k) {
    const float* S = srcs[k] + (long)m * DIM;
    const float* Wk = Wl + k * DIM * DIM;
    for (int j = 0; j < DIM; ++j) acc += S[j] * Wk[j * DIM + n];
  }
  dst[(long)m * DIM + n] = acc > 0.0f ? acc : 0.0f;
}

// ---------------- host orchestration ----------------

extern "C" void kernel_launch(void* const* d_in, const int* in_sizes, int n_in,
                              void* d_out, int out_size, void* d_ws, size_t ws_size,
                              hipStream_t stream) {
  (void)n_in; (void)out_size; (void)ws_size;
  const float* d_x = (const float*)d_in[0];
  const int*   d_ei = (const int*)d_in[1];
  const float* d_w = (const float*)d_in[2];
  const float* d_W = (const float*)d_in[3];
  const float* d_b = (const float*)d_in[4];
  const int N = in_sizes[0] / DIM;
  const int E = in_sizes[2];
  const int* row = d_ei;       // edge_index[0] = sources
  const int* col = d_ei + E;   // edge_index[1] = targets

  auto align_up = [](size_t v) { return (v + 255) & ~(size_t)255; };
  char* ws = (char*)d_ws;
  size_t off = 0;
  float* deg  = (float*)(ws + off); off = align_up(off + (size_t)N * 4);
  float* dis  = (float*)(ws + off); off = align_up(off + (size_t)N * 4);
  float* nrm  = (float*)(ws + off); off = align_up(off + (size_t)E * 4);
  float* h1   = (float*)(ws + off); off = align_up(off + (size_t)N * DIM * 4);
  float* h2   = (float*)(ws + off); off = align_up(off + (size_t)N * DIM * 4);
  float* h3   = (float*)(ws + off); off = align_up(off + (size_t)N * DIM * 4);
  float* xcur = (float*)(ws + off); off = align_up(off + (size_t)N * DIM * 4);

  const int B = 256;
  auto gblk = [&](long n) { return (int)((n + B - 1) / B); };

  // gcn_norm (add_self_loops=False): deg at target, symmetric norm
  zero_f32_kernel<<<gblk(N), B, 0, stream>>>(deg, N);
  deg_kernel<<<gblk(E), B, 0, stream>>>(col, d_w, deg, E);
  dis_kernel<<<gblk(N), B, 0, stream>>>(deg, dis, N);
  norm_kernel<<<gblk(E), B, 0, stream>>>(row, col, d_w, dis, nrm, E);

  const int nTiles = N / 16;
  const int tileBlocks = (nTiles + 7) / 8;   // 8 waves (tiles) per 256-thread block
  const int tailStart = nTiles * 16;
  const int tailRows = N - tailStart;
  const long NF = (long)N * DIM;
  const long hopThreads = (long)E * 32;

  for (int l = 0; l < 3; ++l) {
    const float* xin = (l == 0) ? d_x : xcur;
    float* dst = (l == 2) ? (float*)d_out : xcur;
    const float* Wl = d_W + (size_t)l * 4 * DIM * DIM;
    const float* bl = d_b + (size_t)l * DIM;

    zero_f32_kernel<<<gblk(NF), B, 0, stream>>>(h1, (int)NF);
    hop_kernel<<<gblk(hopThreads), B, 0, stream>>>(row, col, nrm, xin, h1, E);
    zero_f32_kernel<<<gblk(NF), B, 0, stream>>>(h2, (int)NF);
    hop_kernel<<<gblk(hopThreads), B, 0, stream>>>(row, col, nrm, h1, h2, E);
    zero_f32_kernel<<<gblk(NF), B, 0, stream>>>(h3, (int)NF);
    hop_kernel<<<gblk(hopThreads), B, 0, stream>>>(row, col, nrm, h2, h3, E);

    // Fused: dst = relu(bias + xin@W0 + h1@W1 + h2@W2 + h3@W3).
    // In-place for l=1 is safe: each wave reads only its own 16 rows (all reads
    // precede its stores), and tiles are disjoint across waves.
    tag_gemm_wmma_kernel<<<tileBlocks, B, 0, stream>>>(xin, h1, h2, h3, Wl, bl, dst, nTiles);
    if (tailRows > 0) {
      tag_gemm_tail_kernel<<<gblk((long)tailRows * 32), B, 0, stream>>>(
          xin, h1, h2, h3, Wl, bl, dst, tailStart, N);
    }
  }
}